// Biquad_24335284699685
// MI455X (gfx1250) — compile-verified
//
#include <hip/hip_runtime.h>
#include <hip/hip_bf16.h>

// Biquad (DF2T) over [B=512, T=48000] via blocked linear scan + WMMA Toeplitz GEMM.
//
// y_block(16) = L(16x16, lower-tri Toeplitz of impulse response h) * x_block(16)
//             + G(16x2) * s0,   s0 = (z1,z2) at block start.
// L*x for 16 blocks at once = 16x16x16 f32 GEMM = 4 chained v_wmma_f32_16x16x4_f32.
// State carry across blocks recovered from outputs:
//   z2' = b2*x[15] - a2*y[15];  z1' = b1*x[15] - a1*y[15] + (b2*x[14] - a2*y[14])
// Cross-slab initial states via 2-pass chunked scan with propagator M^4000.

typedef __attribute__((ext_vector_type(2))) float v2f;
typedef __attribute__((ext_vector_type(8))) float v8f;

namespace {
constexpr int B_CH = 512;
constexpr int T_LEN = 48000;
constexpr int SLABS = 12;
constexpr int SLAB_LEN = T_LEN / SLABS;               // 4000
constexpr int BLK = 16;                               // samples per block
constexpr int BLOCKS_PER_SLAB = SLAB_LEN / BLK;       // 250
constexpr int GRP = 16;                               // blocks per WMMA group (256 samples)
constexpr int GROUPS = (BLOCKS_PER_SLAB + GRP - 1) / GRP;  // 16 (15 full + tail of 10)
constexpr int COEF_STRIDE = 64;                       // floats per channel in ws
// coef layout per channel: [0..15]=h, [16..31]=gA, [32..47]=gB, [52..55]=M^4000 row-major
}

// ---------------- Kernel A: per-channel coefficient precompute -----------------
__global__ void biquad_coef_kernel(const float* __restrict__ b0v, const float* __restrict__ b1v,
                                   const float* __restrict__ b2v, const float* __restrict__ a1v,
                                   const float* __restrict__ a2v, float* __restrict__ coef) {
  int c = blockIdx.x * blockDim.x + threadIdx.x;
  if (c >= B_CH) return;
  const float b0 = b0v[c], b1 = b1v[c], b2 = b2v[c], a1 = a1v[c], a2 = a2v[c];
  float* w = coef + c * COEF_STRIDE;
  const float v1 = b1 - a1 * b0;  // v = (b1-a1*b0, b2-a2*b0):  s' = M s + v x
  const float v2 = b2 - a2 * b0;
  w[0]  = b0;    // h[0]
  w[16] = 1.0f;  // gA[0]  (g_i = e1^T M^i)
  w[32] = 0.0f;  // gB[0]
  float ga = 1.0f, gb = 0.0f;  // g_{i-1}
  float gaPrevEnd = 0.0f, gbPrevEnd = 0.0f;
  for (int i = 1; i <= SLAB_LEN; ++i) {
    const float gan = -a1 * ga - a2 * gb;  // g_i = g_{i-1} * M
    const float gbn = ga;
    if (i < 16) {
      w[i]      = ga * v1 + gb * v2;  // h[i] = g_{i-1} . v
      w[16 + i] = gan;
      w[32 + i] = gbn;
    }
    if (i == SLAB_LEN - 1) { gaPrevEnd = gan; gbPrevEnd = gbn; }
    ga = gan; gb = gbn;
  }
  // M^SLAB_LEN: row0 = g_SLAB_LEN, row1 = -a2 * g_{SLAB_LEN-1}
  w[52] = ga;            w[53] = gb;
  w[54] = -a2 * gaPrevEnd; w[55] = -a2 * gbPrevEnd;
}

// ---------------- Pass kernel: one wave per (channel, slab) -------------------
template <bool WRITE_Y>
__global__ __launch_bounds__(32) void biquad_pass_kernel(
    const float* __restrict__ x,
    const float* __restrict__ b1v, const float* __restrict__ b2v,
    const float* __restrict__ a1v, const float* __restrict__ a2v,
    const float* __restrict__ coef,
    const float* __restrict__ sstart,   // pass2: [B_CH][SLABS][2]
    float* __restrict__ dloc,           // pass1: [B_CH][SLABS][2]
    float* __restrict__ y) {
  const int lane = threadIdx.x;  // 0..31 (wave32)
  const int wid = blockIdx.x;
  const int c = wid / SLABS;
  const int s = wid % SLABS;
  const int Mrow = lane & 15;    // A-matrix row / C-matrix column index
  const int hi = lane >> 4;
  const float* __restrict__ w = coef + c * COEF_STRIDE;

  // A fragments: A[M][K] = h[M-K] if M>=K else 0; lane=M, K = 4m + r + 2*hi
  v2f af[4];
#pragma unroll
  for (int m = 0; m < 4; ++m) {
#pragma unroll
    for (int r = 0; r < 2; ++r) {
      const int k = 4 * m + r + 2 * hi;
      af[m][r] = (Mrow >= k) ? w[Mrow - k] : 0.0f;
    }
  }
  // Per-lane G rows for output fix-up (this lane owns C rows 8*hi + 0..7)
  float gAr[8], gBr[8];
#pragma unroll
  for (int r = 0; r < 8; ++r) {
    gAr[r] = w[16 + 8 * hi + r];
    gBr[r] = w[32 + 8 * hi + r];
  }
  const float gA14 = w[16 + 14], gA15 = w[16 + 15];
  const float gB14 = w[32 + 14], gB15 = w[32 + 15];
  const float b1 = b1v[c], b2 = b2v[c], a1 = a1v[c], a2 = a2v[c];

  float z1 = 0.0f, z2 = 0.0f;
  if constexpr (WRITE_Y) {
    z1 = sstart[(c * SLABS + s) * 2 + 0];
    z2 = sstart[(c * SLABS + s) * 2 + 1];
  }

  const long base_ch = (long)c * T_LEN + (long)s * SLAB_LEN;

  for (int G = 0; G < GROUPS; ++G) {
    const int j0 = G * GRP;
    const int nb = min(GRP, BLOCKS_PER_SLAB - j0);  // 16, tail = 10
    const long base = base_ch + (long)j0 * BLK;

    // B fragments: B[K][N] = x[base + N*16 + K]; lane = N (= Mrow), K = 4m + r + 2*hi
    v2f bf[4];
#pragma unroll
    for (int m = 0; m < 4; ++m) {
      if (Mrow < nb) {
        const float2 t = *(const float2*)(x + base + Mrow * BLK + 4 * m + 2 * hi);
        bf[m][0] = t.x; bf[m][1] = t.y;
      } else {
        bf[m][0] = 0.0f; bf[m][1] = 0.0f;
      }
    }

    // Prefetch next group's 1KB (speculative; dropped on translation failure)
    if (G + 1 < GROUPS)
      __builtin_prefetch((const void*)(x + base + GRP * BLK + lane * 8), 0, 1);

    // Y_part(16x16) = L * X  via 4 chained K=4 WMMAs
    v8f acc = {};
    acc = __builtin_amdgcn_wmma_f32_16x16x4_f32(false, af[0], false, bf[0], (short)0, acc, false, false);
    acc = __builtin_amdgcn_wmma_f32_16x16x4_f32(false, af[1], false, bf[1], (short)0, acc, false, false);
    acc = __builtin_amdgcn_wmma_f32_16x16x4_f32(false, af[2], false, bf[2], (short)0, acc, false, false);
    acc = __builtin_amdgcn_wmma_f32_16x16x4_f32(false, af[3], false, bf[3], (short)0, acc, false, false);

    // Serial cross-block state fix-up (16 cheap steps instead of 256)
    for (int jj = 0; jj < nb; ++jj) {
      const int src = 16 + jj;  // hi lane jj holds rows 14,15 of column jj
      const float yp14 = __shfl(acc[6], src, 32);
      const float yp15 = __shfl(acc[7], src, 32);
      const float x14 = __shfl(bf[3][0], src, 32);  // hi-lane bf[3] = x[k=14], x[k=15]
      const float x15 = __shfl(bf[3][1], src, 32);
      const float y14 = yp14 + gA14 * z1 + gB14 * z2;
      const float y15 = yp15 + gA15 * z1 + gB15 * z2;
      if constexpr (WRITE_Y) {
        if (Mrow == jj) {
#pragma unroll
          for (int r = 0; r < 8; ++r) acc[r] += gAr[r] * z1 + gBr[r] * z2;
        }
      }
      const float z2mid = b2 * x14 - a2 * y14;  // z2 entering sample 15
      const float z2n = b2 * x15 - a2 * y15;
      const float z1n = b1 * x15 - a1 * y15 + z2mid;
      z1 = z1n; z2 = z2n;
    }

    if constexpr (WRITE_Y) {
      if (Mrow < nb) {
        float* dst = y + base + Mrow * BLK + 8 * hi;  // column Mrow, rows 8*hi..8*hi+7
        *(float4*)(dst + 0) = make_float4(acc[0], acc[1], acc[2], acc[3]);
        *(float4*)(dst + 4) = make_float4(acc[4], acc[5], acc[6], acc[7]);
      }
    }
  }

  if constexpr (!WRITE_Y) {
    if (lane == 0) {
      dloc[(c * SLABS + s) * 2 + 0] = z1;
      dloc[(c * SLABS + s) * 2 + 1] = z2;
    }
  }
}

// ---------------- Kernel C: tiny per-channel scan over slabs ------------------
__global__ void biquad_scan_kernel(const float* __restrict__ coef,
                                   const float* __restrict__ dloc,
                                   float* __restrict__ sstart) {
  int c = blockIdx.x * blockDim.x + threadIdx.x;
  if (c >= B_CH) return;
  const float* w = coef + c * COEF_STRIDE;
  const float m00 = w[52], m01 = w[53], m10 = w[54], m11 = w[55];
  float z1 = 0.0f, z2 = 0.0f;
  for (int s = 0; s < SLABS; ++s) {
    sstart[(c * SLABS + s) * 2 + 0] = z1;
    sstart[(c * SLABS + s) * 2 + 1] = z2;
    const float d1 = dloc[(c * SLABS + s) * 2 + 0];
    const float d2 = dloc[(c * SLABS + s) * 2 + 1];
    const float z1n = m00 * z1 + m01 * z2 + d1;  // s_{k+1} = M^4000 s_k + d_k
    const float z2n = m10 * z1 + m11 * z2 + d2;
    z1 = z1n; z2 = z2n;
  }
}

extern "C" void kernel_launch(void* const* d_in, const int* in_sizes, int n_in,
                              void* d_out, int out_size, void* d_ws, size_t ws_size,
                              hipStream_t stream) {
  const float* x  = (const float*)d_in[0];
  const float* b0 = (const float*)d_in[1];
  const float* b1 = (const float*)d_in[2];
  const float* b2 = (const float*)d_in[3];
  const float* a1 = (const float*)d_in[4];
  const float* a2 = (const float*)d_in[5];
  float* y = (float*)d_out;

  float* ws     = (float*)d_ws;
  float* coef   = ws;                        // 512*64 floats  = 128 KB
  float* dloc   = coef + B_CH * COEF_STRIDE; // 512*12*2 floats = 48 KB
  float* sstart = dloc + B_CH * SLABS * 2;   // 512*12*2 floats = 48 KB

  biquad_coef_kernel<<<(B_CH + 63) / 64, 64, 0, stream>>>(b0, b1, b2, a1, a2, coef);
  biquad_pass_kernel<false><<<B_CH * SLABS, 32, 0, stream>>>(
      x, b1, b2, a1, a2, coef, nullptr, dloc, nullptr);
  biquad_scan_kernel<<<(B_CH + 63) / 64, 64, 0, stream>>>(coef, dloc, sstart);
  biquad_pass_kernel<true><<<B_CH * SLABS, 32, 0, stream>>>(
      x, b1, b2, a1, a2, coef, sstart, nullptr, y);
}